// SAMPONet_60756607369494
// MI455X (gfx1250) — compile-verified
//
#include <hip/hip_runtime.h>

#define Bb 64
#define Nn 2048
#define Hh 512
#define NF 256

typedef float v2f __attribute__((ext_vector_type(2)));
typedef float v8f __attribute__((ext_vector_type(8)));

__device__ __forceinline__ v8f wmma_f32(v2f a, v2f b, v8f c) {
  // (neg_a, A, neg_b, B, c_mod, C, reuse_a, reuse_b)
  return __builtin_amdgcn_wmma_f32_16x16x4_f32(false, a, false, b, (short)0, c,
                                               false, false);
}

// xT[h][b] = vs[b][h]  (transposed copy so WMMA B-operand loads are lane-contiguous)
__global__ void k_xT(const float* __restrict__ vs, float* __restrict__ xT) {
  int h = threadIdx.x, b = blockIdx.x;
  xT[h * 64 + b] = vs[b * 512 + h];
}

// biasQ[b][k] = sum_h x[b,h] * bq[h*512+k]
__global__ void k_biasQ(const float* __restrict__ vs, const float* __restrict__ bq,
                        float* __restrict__ biasQ) {
  int b = blockIdx.x;
  int k = blockIdx.y * 128 + threadIdx.x;
  float acc = 0.f;
  for (int h = 0; h < 512; ++h) acc += vs[b * 512 + h] * bq[h * 512 + k];
  biasQ[b * 512 + k] = acc;
}

// GEMM1: Qpart[pc][k][b] += sum_{p in chunk} Wq2[p,k] * (x[b,j]*x[b,h]), p=(j,h)
// WMMA axes: M = k (16), N = b (16, x4 tiles), K = p (4)
__global__ void __launch_bounds__(32) k_gemm1(const float* __restrict__ Wq,
                                              const float* __restrict__ xT,
                                              float* __restrict__ Qpart) {
  const int lane  = threadIdx.x;
  const int ktile = blockIdx.x;   // 0..31
  const int pc    = blockIdx.y;   // 0..127 -> 4 j's each
  const int half  = lane >> 4;
  const int bl    = lane & 15;
  const int kg    = ktile * 16 + (lane & 15);  // A-row (output k)
  const int c0    = half * 2;

  v8f acc[4];
  for (int bt = 0; bt < 4; ++bt)
    for (int i = 0; i < 8; ++i) acc[bt][i] = 0.f;

  const int j0 = pc * 4;
  for (int jj = 0; jj < 4; ++jj) {
    const int j = j0 + jj;
    float xj[4];
    for (int bt = 0; bt < 4; ++bt) xj[bt] = xT[j * 64 + bt * 16 + bl];
    const float* wrow = Wq + (size_t)(j * 512) * 512 + kg;
    if (jj < 3) __builtin_prefetch(Wq + (size_t)((j + 1) * 512) * 512 + kg, 0, 0);
#pragma unroll 2
    for (int hb = 0; hb < 512; hb += 4) {
      const int h0 = hb + c0;
      v2f A;
      A.x = wrow[(size_t)h0 * 512];
      A.y = wrow[(size_t)(h0 + 1) * 512];
      for (int bt = 0; bt < 4; ++bt) {
        v2f Bm;
        Bm.x = xT[h0 * 64 + bt * 16 + bl] * xj[bt];
        Bm.y = xT[(h0 + 1) * 64 + bt * 16 + bl] * xj[bt];
        acc[bt] = wmma_f32(A, Bm, acc[bt]);
      }
    }
  }
  float* qp = Qpart + (size_t)pc * 512 * 64;
  for (int bt = 0; bt < 4; ++bt)
    for (int i = 0; i < 8; ++i) {
      int krow = ktile * 16 + i + 8 * half;   // D: M = i + 8*half
      qp[krow * 64 + bt * 16 + bl] = acc[bt][i];
    }
}

// QT[k][b] = biasQ[b][k] + sum_pc Qpart[pc][k][b]
__global__ void k_qreduce(const float* __restrict__ Qpart,
                          const float* __restrict__ biasQ, float* __restrict__ QT) {
  int k = blockIdx.x, b = threadIdx.x;
  float acc = biasQ[b * 512 + k];
  for (int pc = 0; pc < 128; ++pc) acc += Qpart[((size_t)pc * 512 + k) * 64 + b];
  QT[k * 64 + b] = acc;
}

// tbias[b][h] = sum_k Q[b,k]*bk[h*512+k]
__global__ void k_tbias(const float* __restrict__ QT, const float* __restrict__ bk,
                        float* __restrict__ tbias) {
  int b = blockIdx.x;
  int h = blockIdx.y * 128 + threadIdx.x;
  float acc = 0.f;
  for (int k = 0; k < 512; ++k) acc += QT[k * 64 + b] * bk[h * 512 + k];
  tbias[b * 512 + h] = acc;
}

// GEMM2: for j in group: C[r=(j,h), b] = sum_k Wk2[r,k]*Q[b,k]; fold x[b,j] into tpart
// WMMA axes: M = h-row (16), N = b (16, x4 tiles), K = k (4)
__global__ void __launch_bounds__(32) k_gemm2(const float* __restrict__ Wk,
                                              const float* __restrict__ QT,
                                              const float* __restrict__ xT,
                                              float* __restrict__ tpart) {
  const int lane  = threadIdx.x;
  const int htile = blockIdx.x;   // 0..31
  const int jg    = blockIdx.y;   // 0..63 -> 8 j's each
  const int half  = lane >> 4;
  const int bl    = lane & 15;
  const int c0    = half * 2;

  v8f tacc[4];
  for (int bt = 0; bt < 4; ++bt)
    for (int i = 0; i < 8; ++i) tacc[bt][i] = 0.f;

  for (int jj = 0; jj < 8; ++jj) {
    const int j = jg * 8 + jj;
    const float* arow =
        Wk + (size_t)(j * 512 + htile * 16 + (lane & 15)) * 512;
    if (jj < 7)
      __builtin_prefetch(
          Wk + (size_t)((j + 1) * 512 + htile * 16 + (lane & 15)) * 512, 0, 0);
    v8f c[4];
    for (int bt = 0; bt < 4; ++bt)
      for (int i = 0; i < 8; ++i) c[bt][i] = 0.f;
#pragma unroll 2
    for (int kb = 0; kb < 512; kb += 4) {
      v2f A;
      A.x = arow[kb + c0];
      A.y = arow[kb + c0 + 1];
      for (int bt = 0; bt < 4; ++bt) {
        v2f Bm;
        Bm.x = QT[(kb + c0) * 64 + bt * 16 + bl];
        Bm.y = QT[(kb + c0 + 1) * 64 + bt * 16 + bl];
        c[bt] = wmma_f32(A, Bm, c[bt]);
      }
    }
    for (int bt = 0; bt < 4; ++bt) {
      float xjb = xT[j * 64 + bt * 16 + bl];
      for (int i = 0; i < 8; ++i) tacc[bt][i] += c[bt][i] * xjb;
    }
  }
  float* tp = tpart + (size_t)jg * 512 * 64;
  for (int bt = 0; bt < 4; ++bt)
    for (int i = 0; i < 8; ++i) {
      int hrow = htile * 16 + i + 8 * half;
      tp[hrow * 64 + bt * 16 + bl] = tacc[bt][i];
    }
}

// t[b][h] = tbias[b][h] + sum_jg tpart[jg][h][b]
__global__ void k_treduce(const float* __restrict__ tpart,
                          const float* __restrict__ tbias, float* __restrict__ t) {
  int h = blockIdx.x, b = threadIdx.x;
  float acc = tbias[b * 512 + h];
  for (int jg = 0; jg < 64; ++jg) acc += tpart[((size_t)jg * 512 + h) * 64 + b];
  t[b * 512 + h] = acc;
}

// compat[b][n] = (1/sqrt(H)) * sum_h ve[b,n,h]*t[b,h]
__global__ void __launch_bounds__(256) k_compat(const float* __restrict__ ve,
                                                const float* __restrict__ t,
                                                float* __restrict__ compat) {
  __shared__ float ts[512];
  int b = blockIdx.x;
  for (int i = threadIdx.x; i < 512; i += 256) ts[i] = t[b * 512 + i];
  __syncthreads();
  int wave = threadIdx.x >> 5, lane = threadIdx.x & 31;
  int n = blockIdx.y * 8 + wave;
  const float* row = ve + ((size_t)b * 2048 + n) * 512;
  float acc = 0.f;
#pragma unroll
  for (int i = 0; i < 16; ++i) {
    int h = lane + 32 * i;
    acc += row[h] * ts[h];
  }
  for (int off = 16; off > 0; off >>= 1) acc += __shfl_xor(acc, off, 32);
  if (lane == 0) compat[b * 2048 + n] = acc * 0.04419417382415922f;
}

// Per-batch: bitonic sort 2048 (val,idx) desc (tie -> idx asc), gather top-256 rows
__global__ void __launch_bounds__(1024) k_topk(const float* __restrict__ compat,
                                               const float* __restrict__ ve,
                                               float* __restrict__ out) {
  __shared__ float sv[2048];
  __shared__ int si[2048];
  int b = blockIdx.x, tid = threadIdx.x;
  for (int i = tid; i < 2048; i += 1024) {
    sv[i] = compat[b * 2048 + i];
    si[i] = i;
  }
  __syncthreads();
  for (int ksz = 2; ksz <= 2048; ksz <<= 1) {
    for (int stride = ksz >> 1; stride > 0; stride >>= 1) {
      for (int e = tid; e < 2048; e += 1024) {
        int ixj = e ^ stride;
        if (ixj > e) {
          float v0 = sv[e], v1 = sv[ixj];
          int i0 = si[e], i1 = si[ixj];
          bool descending = ((e & ksz) == 0);
          bool firstAfter = (v0 < v1) || (v0 == v1 && i0 > i1);
          if (firstAfter == descending) {
            sv[e] = v1; sv[ixj] = v0;
            si[e] = i1; si[ixj] = i0;
          }
        }
      }
      __syncthreads();
    }
  }
  const float4* vev = (const float4*)ve;
  float4* outv = (float4*)out;
  for (int e = tid; e < 256 * 128; e += 1024) {
    int r = e >> 7, cix = e & 127;
    int src = si[r];
    outv[((size_t)b * 256 + r) * 128 + cix] =
        vev[((size_t)b * 2048 + src) * 128 + cix];
  }
}

extern "C" void kernel_launch(void* const* d_in, const int* in_sizes, int n_in,
                              void* d_out, int out_size, void* d_ws, size_t ws_size,
                              hipStream_t stream) {
  const float* vs = (const float*)d_in[0];
  const float* ve = (const float*)d_in[1];
  const float* Wq = (const float*)d_in[2];
  const float* bq = (const float*)d_in[3];
  const float* Wk = (const float*)d_in[4];
  const float* bk = (const float*)d_in[5];
  float* out = (float*)d_out;
  float* ws = (float*)d_ws;

  float* xT     = ws;            // 32768 floats
  float* biasQ  = ws + 32768;    // 32768
  float* QT     = ws + 65536;    // 32768
  float* tbias  = ws + 98304;    // 32768
  float* t      = ws + 131072;   // 32768
  float* compat = ws + 163840;   // 131072
  float* Qpart  = ws + 294912;   // 128*512*64 = 4194304
  float* tpart  = ws + 4489216;  // 64*512*64  = 2097152
  // total ~26.3 MB of workspace

  k_xT<<<64, 512, 0, stream>>>(vs, xT);
  k_biasQ<<<dim3(64, 4), 128, 0, stream>>>(vs, bq, biasQ);
  k_gemm1<<<dim3(32, 128), 32, 0, stream>>>(Wq, xT, Qpart);
  k_qreduce<<<512, 64, 0, stream>>>(Qpart, biasQ, QT);
  k_tbias<<<dim3(64, 4), 128, 0, stream>>>(QT, bk, tbias);
  k_gemm2<<<dim3(32, 64), 32, 0, stream>>>(Wk, QT, xT, tpart);
  k_treduce<<<512, 64, 0, stream>>>(tpart, tbias, t);
  k_compat<<<dim3(64, 256), 256, 0, stream>>>(ve, t, compat);
  k_topk<<<64, 1024, 0, stream>>>(compat, ve, out);
}